// SoftMaskingModule_21689584845637
// MI455X (gfx1250) — compile-verified
//
#include <hip/hip_runtime.h>
#include <math.h>

// Problem constants (from reference)
#define BATCH   2
#define SEQ     2048
#define VOCAB   50257
#define HID     1024
#define KTOP    5
#define MASK_ID 50256

#define BDIM    256           // threads per block (8 waves of 32)
#define CHUNK   4096          // floats per LDS staging buffer (16 KB)
// per-chunk: each thread issues CHUNK/(BDIM*4) = 4 b128 async loads

#define LN2F    0.69314718055994530942f

__device__ __forceinline__ void topk_insert(float v[KTOP], int ix[KTOP], float p, int idx) {
  // strict '>' keeps earliest index on ties (per-thread scan order is increasing index)
  if (p > v[KTOP - 1]) {
    v[KTOP - 1] = p; ix[KTOP - 1] = idx;
#pragma unroll
    for (int k = KTOP - 1; k > 0; --k) {
      if (v[k] > v[k - 1]) {
        float tv = v[k]; v[k] = v[k - 1]; v[k - 1] = tv;
        int   ti = ix[k]; ix[k] = ix[k - 1]; ix[k - 1] = ti;
      }
    }
  }
}

// ent2 accumulates sum(p * log2(p)); entropy = -ln2 * ent2.
// fmax(p, 1e-45) makes p==0 contribute exactly 0 (0 * -149.2) => matches
// jnp.where(p>0, -p*log(p), 0) without a compare/select. v_log_f32 is TRANS32.
__device__ __forceinline__ void process_elem(float p, int idx, float& ent2,
                                             float v[KTOP], int ix[KTOP]) {
  ent2 = fmaf(p, __log2f(fmaxf(p, 1e-45f)), ent2);
  topk_insert(v, ix, p, idx);
}

__global__ __launch_bounds__(BDIM) void softmask_kernel(
    const int*   __restrict__ x_t,
    const float* __restrict__ probs,
    const float* __restrict__ emb,
    const float* __restrict__ p_os,
    const float* __restrict__ p_oa,
    const float* __restrict__ p_ob,
    float*       __restrict__ out)
{
  const int row = blockIdx.x;     // 0 .. B*S-1
  const int tid = threadIdx.x;

  const int xt = x_t[row];
  float* orow = out + (size_t)row * HID;

  // ---------- fast path: real token, just copy its embedding ----------
  if (xt != MASK_ID) {
    const float4* e4 = (const float4*)(emb + (size_t)xt * HID);
    float4*       o4 = (float4*)orow;
    for (int h = tid; h < HID / 4; h += BDIM) o4[h] = e4[h];
    return;
  }

  // ---------- masked token: scan prob row (entropy + top-5) ----------
  __shared__ __align__(16) float s_buf[2][CHUNK];   // async staging (32 KB)
  __shared__ float s_val[BDIM * KTOP];
  __shared__ int   s_idx[BDIM * KTOP];
  __shared__ float s_ent[BDIM];
  __shared__ float s_coef[KTOP + 1];
  __shared__ int   s_ci[KTOP];

  const float* prow = probs + (size_t)row * VOCAB;

  float tv[KTOP]; int ti[KTOP];
#pragma unroll
  for (int k = 0; k < KTOP; ++k) { tv[k] = -1.0f; ti[k] = 0x7fffffff; }
  float ent2 = 0.0f;   // sum p*log2(p)

  // prologue until 16-byte alignment (row base is only 4B aligned: V is odd)
  const int pro = (int)(((16u - ((uint32_t)(uintptr_t)prow & 15u)) & 15u) >> 2);
  for (int i = tid; i < pro; i += BDIM) {
    process_elem(prow[i], i, ent2, tv, ti);
  }

  const float* pal     = prow + pro;
  const int    nmain   = VOCAB - pro;
  const int    nchunks = nmain / CHUNK;

  // issue one chunk: per-lane 16B async copies, global -> LDS (ASYNCcnt)
  auto issue = [&](int c) {
    const uint32_t lds0 = (uint32_t)(uintptr_t)(&s_buf[c & 1][0]);
    const char*    g    = (const char*)(pal + (size_t)c * CHUNK);
#pragma unroll
    for (int u = 0; u < CHUNK / (BDIM * 4); ++u) {
      uint32_t    loff = lds0 + (uint32_t)((tid + u * BDIM) * 16);
      const void* ga   = (const void*)(g + (size_t)(tid + u * BDIM) * 16);
      asm volatile("global_load_async_to_lds_b128 %0, %1, off"
                   :: "v"(loff), "v"(ga) : "memory");
    }
  };

  if (nchunks > 0) issue(0);
  for (int c = 0; c < nchunks; ++c) {
    if (c + 1 < nchunks) {
      issue(c + 1);                                       // double-buffer ahead
      asm volatile("s_wait_asynccnt 4" ::: "memory");     // oldest 4 (chunk c) done
    } else {
      asm volatile("s_wait_asynccnt 0" ::: "memory");
    }
    __syncthreads();                                      // all waves' slices landed

    // process buffer via ds_load_b128: 4 floats per DS op, contiguous per lane.
    // per-thread global index order stays strictly increasing (tie-break safe).
    const float4* bp4   = (const float4*)(&s_buf[c & 1][0]);
    const int     gbase = pro + c * CHUNK;
#pragma unroll
    for (int u = 0; u < CHUNK / (BDIM * 4); ++u) {
      const int    q  = tid + u * BDIM;       // float4 slot
      const float4 pv = bp4[q];
      const int    gi = gbase + q * 4;
      process_elem(pv.x, gi + 0, ent2, tv, ti);
      process_elem(pv.y, gi + 1, ent2, tv, ti);
      process_elem(pv.z, gi + 2, ent2, tv, ti);
      process_elem(pv.w, gi + 3, ent2, tv, ti);
    }
    __syncthreads();                                      // done reading before overwrite
  }

  // scalar tail
  for (int i = pro + nchunks * CHUNK + tid; i < VOCAB; i += BDIM) {
    process_elem(prow[i], i, ent2, tv, ti);
  }

  // ---------- block reduction: entropy sum + top-5 merge ----------
#pragma unroll
  for (int k = 0; k < KTOP; ++k) { s_val[tid * KTOP + k] = tv[k]; s_idx[tid * KTOP + k] = ti[k]; }
  s_ent[tid] = -LN2F * ent2;   // convert sum(p*log2 p) -> -sum(p*ln p)
  __syncthreads();

  for (int s = BDIM / 2; s > 0; s >>= 1) {
    if (tid < s) {
      s_ent[tid] += s_ent[tid + s];
      float a0 = s_val[tid*KTOP+0], a1 = s_val[tid*KTOP+1], a2 = s_val[tid*KTOP+2],
            a3 = s_val[tid*KTOP+3], a4 = s_val[tid*KTOP+4];
      int   ia0 = s_idx[tid*KTOP+0], ia1 = s_idx[tid*KTOP+1], ia2 = s_idx[tid*KTOP+2],
            ia3 = s_idx[tid*KTOP+3], ia4 = s_idx[tid*KTOP+4];
      float b0 = s_val[(tid+s)*KTOP+0], b1 = s_val[(tid+s)*KTOP+1], b2 = s_val[(tid+s)*KTOP+2],
            b3 = s_val[(tid+s)*KTOP+3], b4 = s_val[(tid+s)*KTOP+4];
      int   ib0 = s_idx[(tid+s)*KTOP+0], ib1 = s_idx[(tid+s)*KTOP+1], ib2 = s_idx[(tid+s)*KTOP+2],
            ib3 = s_idx[(tid+s)*KTOP+3], ib4 = s_idx[(tid+s)*KTOP+4];
      float ov[KTOP]; int oi[KTOP];
#pragma unroll
      for (int m = 0; m < KTOP; ++m) {
        bool takeA = (a0 > b0) || ((a0 == b0) && (ia0 <= ib0));  // lax.top_k tie-break
        ov[m] = takeA ? a0 : b0;
        oi[m] = takeA ? ia0 : ib0;
        if (takeA) { a0=a1; a1=a2; a2=a3; a3=a4; a4=-2.0f; ia0=ia1; ia1=ia2; ia2=ia3; ia3=ia4; }
        else       { b0=b1; b1=b2; b2=b3; b3=b4; b4=-2.0f; ib0=ib1; ib1=ib2; ib2=ib3; ib3=ib4; }
      }
#pragma unroll
      for (int k = 0; k < KTOP; ++k) { s_val[tid*KTOP+k] = ov[k]; s_idx[tid*KTOP+k] = oi[k]; }
    }
    __syncthreads();
  }

  // ---------- lambda + blend coefficients (thread 0) ----------
  if (tid == 0) {
    float wsum = s_val[0] + s_val[1] + s_val[2] + s_val[3] + s_val[4] + 1e-10f;
    float Hent = s_ent[0];
    float os = *p_os, oa = *p_oa, ob = *p_ob;
    float real_s = 1.0f / (1.0f + expf(-os));
    float sp_a = (oa > 20.0f) ? oa : log1pf(expf(oa));    // softplus(omega_a)
    float sp_b = (ob > 20.0f) ? ob : log1pf(expf(ob));    // softplus(omega_b)
    float z    = sp_a * (-Hent + sp_b);                   // real_b = -softplus(ob)
    float lam  = real_s / (1.0f + expf(-z));
    s_coef[KTOP] = 1.0f - lam;
#pragma unroll
    for (int k = 0; k < KTOP; ++k) { s_coef[k] = lam * s_val[k] / wsum; s_ci[k] = s_idx[k]; }
  }
  __syncthreads();

  // ---------- gather-blend output: lam*sum(w_k e_k) + (1-lam)*mask ----------
  const float c0 = s_coef[0], c1 = s_coef[1], c2 = s_coef[2],
              c3 = s_coef[3], c4 = s_coef[4], cm = s_coef[KTOP];
  const float4* e0 = (const float4*)(emb + (size_t)s_ci[0] * HID);
  const float4* e1 = (const float4*)(emb + (size_t)s_ci[1] * HID);
  const float4* e2 = (const float4*)(emb + (size_t)s_ci[2] * HID);
  const float4* e3 = (const float4*)(emb + (size_t)s_ci[3] * HID);
  const float4* e4 = (const float4*)(emb + (size_t)s_ci[4] * HID);
  const float4* em = (const float4*)(emb + (size_t)MASK_ID * HID);
  float4* o4 = (float4*)orow;
  for (int h = tid; h < HID / 4; h += BDIM) {
    float4 v0 = e0[h], v1 = e1[h], v2 = e2[h], v3 = e3[h], v4 = e4[h], vm = em[h];
    float4 r;
    r.x = c0*v0.x + c1*v1.x + c2*v2.x + c3*v3.x + c4*v4.x + cm*vm.x;
    r.y = c0*v0.y + c1*v1.y + c2*v2.y + c3*v3.y + c4*v4.y + cm*vm.y;
    r.z = c0*v0.z + c1*v1.z + c2*v2.z + c3*v3.z + c4*v4.z + cm*vm.z;
    r.w = c0*v0.w + c1*v1.w + c2*v2.w + c3*v3.w + c4*v4.w + cm*vm.w;
    o4[h] = r;
  }
}

extern "C" void kernel_launch(void* const* d_in, const int* in_sizes, int n_in,
                              void* d_out, int out_size, void* d_ws, size_t ws_size,
                              hipStream_t stream) {
  (void)in_sizes; (void)n_in; (void)out_size; (void)d_ws; (void)ws_size;
  const int*   x_t   = (const int*)d_in[0];
  const float* probs = (const float*)d_in[1];
  const float* emb   = (const float*)d_in[2];
  const float* os    = (const float*)d_in[3];
  const float* oa    = (const float*)d_in[4];
  const float* ob    = (const float*)d_in[5];
  float*       out   = (float*)d_out;

  dim3 grid(BATCH * SEQ);
  dim3 block(BDIM);
  hipLaunchKernelGGL(softmask_kernel, grid, block, 0, stream,
                     x_t, probs, emb, os, oa, ob, out);
}